// AtomicConvolution_498216207041
// MI455X (gfx1250) — compile-verified
//
#include <hip/hip_runtime.h>

typedef __attribute__((ext_vector_type(2))) float v2f;
typedef __attribute__((ext_vector_type(8))) float v8f;

#define BB 64
#define NN 1024
#define MM 64
#define CC 48
#define PI_F 3.14159265358979323846f

// One wave32 per (b,n) site. Per site: compute R/Z for 64 neighbors (2 per lane),
// then 16 x V_WMMA_F32_16X16X4_F32 accumulate sym = rsf^T(12x64) @ cond(64x4),
// padded to 16x16. Full fp32 precision end-to-end.
__global__ __launch_bounds__(256)
void acnn_sym_kernel(const float* __restrict__ X,
                     const float* __restrict__ rp,
                     const int*   __restrict__ Nbrs,
                     const int*   __restrict__ NbrsZ,
                     float*       __restrict__ out)
{
    const int lane = threadIdx.x & 31;
    const int wave = threadIdx.x >> 5;
    const int site = blockIdx.x * 8 + wave;     // 0 .. B*N-1 (grid sized exactly)
    const int b    = site >> 10;                // site / NN
    // center coordinates (uniform across wave)
    const float cx = X[(long)site * 3 + 0];
    const float cy = X[(long)site * 3 + 1];
    const float cz = X[(long)site * 3 + 2];

    // Each lane handles neighbors m0 = 2*lane, m1 = 2*lane+1
    float R0, R1;
    int   Z0, Z1;
    {
        const long nb = (long)site * MM + 2 * lane;
        const int i0 = Nbrs[nb + 0];
        const int i1 = Nbrs[nb + 1];
        Z0 = NbrsZ[nb + 0];
        Z1 = NbrsZ[nb + 1];
        const long p0 = ((long)b * NN + i0) * 3;
        const long p1 = ((long)b * NN + i1) * 3;
        float dx = X[p0 + 0] - cx, dy = X[p0 + 1] - cy, dz = X[p0 + 2] - cz;
        R0 = sqrtf(dx * dx + dy * dy + dz * dz);
        dx = X[p1 + 0] - cx; dy = X[p1 + 1] - cy; dz = X[p1 + 2] - cz;
        R1 = sqrtf(dx * dx + dy * dy + dz * dz);
    }

    // A-matrix row for this lane (ISA 32-bit A 16x4 layout: M = lane&15 for both halves,
    // K = {0,1} for lanes 0-15, K = {2,3} for lanes 16-31)
    const int  row      = lane & 15;
    const int  hi       = lane >> 4;
    const bool rowValid = (row < 12);
    const int  l        = rowValid ? row : 0;
    const float rc    = rp[l * 3 + 0];
    const float rs    = rp[l * 3 + 1];
    const float ee    = rp[l * 3 + 2];
    const float pi_rc = PI_F / rc;

    // B-matrix column atom type (cols 0..3 = {1,6,7,8}; Z in [0,10) never equals -1)
    const int myType = (row == 0) ? 1 : (row == 1) ? 6 : (row == 2) ? 7 : (row == 3) ? 8 : -1;

    v8f acc = {};
#pragma unroll
    for (int k = 0; k < 16; ++k) {
        // neighbor m = 4k + (hi?2:0) + {0,1} lives in lane 2k+hi as its (R0,R1)/(Z0,Z1)
        const int src = 2 * k + hi;
        const float Ra = __shfl(R0, src, 32);
        const float Rb = __shfl(R1, src, 32);
        const int   Za = __shfl(Z0, src, 32);
        const int   Zb = __shfl(Z1, src, 32);

        const float d0 = Ra - rs;
        const float d1 = Rb - rs;
        const float k0 = __expf(-ee * d0 * d0);
        const float k1 = __expf(-ee * d1 * d1);
        const float t0 = 0.5f * (__cosf(pi_rc * Ra) + 1.0f);
        const float t1 = 0.5f * (__cosf(pi_rc * Rb) + 1.0f);
        const float f0 = (Ra <= rc) ? t0 : 0.0f;
        const float f1 = (Rb <= rc) ? t1 : 0.0f;

        v2f a;
        a.x = rowValid ? k0 * f0 : 0.0f;
        a.y = rowValid ? k1 * f1 : 0.0f;
        v2f bmat;
        bmat.x = (Za == myType) ? 1.0f : 0.0f;
        bmat.y = (Zb == myType) ? 1.0f : 0.0f;

        acc = __builtin_amdgcn_wmma_f32_16x16x4_f32(
            /*neg_a=*/false, a, /*neg_b=*/false, bmat,
            /*c_mod=*/(short)0, acc, /*reuse_a=*/false, /*reuse_b=*/false);
    }

    // C/D 16x16 f32 layout: VGPR r holds M=r (lanes 0-15, N=lane) and M=r+8 (lanes 16-31, N=lane-16).
    // sym channel c = l*4 + a; valid l in 0..11, a in 0..3.
    const long base = (long)site * CC;
    if (row < 4) {
        if (hi == 0) {
#pragma unroll
            for (int r = 0; r < 8; ++r)
                out[base + r * 4 + row] = acc[r];          // l = r, a = row
        } else {
#pragma unroll
            for (int r = 0; r < 4; ++r)
                out[base + (r + 8) * 4 + row] = acc[r];    // l = r+8, a = row
        }
    }
}

// Batch-norm over batch axis: one thread per (n,c) column; in-place on d_out.
__global__ __launch_bounds__(256)
void acnn_bn_kernel(float* __restrict__ out)
{
    const int idx = blockIdx.x * blockDim.x + threadIdx.x;   // (n,c) flattened
    if (idx >= NN * CC) return;
    const long stride = (long)NN * CC;

    float s = 0.0f, s2 = 0.0f;
#pragma unroll 4
    for (int bb = 0; bb < BB; ++bb) {
        const float v = out[(long)bb * stride + idx];
        s  += v;
        s2 += v * v;
    }
    const float mean = s * (1.0f / BB);
    const float var  = s2 * (1.0f / BB) - mean * mean;   // population variance (jnp.var)
    const float inv  = rsqrtf(var + 0.001f);             // BN_EPS

#pragma unroll 4
    for (int bb = 0; bb < BB; ++bb) {
        const long o = (long)bb * stride + idx;
        out[o] = (out[o] - mean) * inv;
    }
}

extern "C" void kernel_launch(void* const* d_in, const int* in_sizes, int n_in,
                              void* d_out, int out_size, void* d_ws, size_t ws_size,
                              hipStream_t stream) {
    (void)in_sizes; (void)n_in; (void)d_ws; (void)ws_size; (void)out_size;
    const float* X     = (const float*)d_in[0];
    const float* rp    = (const float*)d_in[1];
    const int*   Nbrs  = (const int*)d_in[2];
    const int*   NbrsZ = (const int*)d_in[3];
    float*       out   = (float*)d_out;

    // Kernel 1: 64*1024 sites, 8 waves (one site each) per 256-thread block.
    dim3 block(256);
    dim3 grid1((BB * NN) / 8);
    acnn_sym_kernel<<<grid1, block, 0, stream>>>(X, rp, Nbrs, NbrsZ, out);

    // Kernel 2: batch-norm over batch axis, in place.
    dim3 grid2((NN * CC + 255) / 256);
    acnn_bn_kernel<<<grid2, block, 0, stream>>>(out);
}